// LightGCN_34187939676701
// MI455X (gfx1250) — compile-verified
//
#include <hip/hip_runtime.h>

#define CHUNK 128
#define TPB 256
#define SCATTER_BLOCKS 1024

typedef int v4i_gcc __attribute__((vector_size(16)));

// ---------------------------------------------------------------------------
// CDNA5 async global->LDS path (probe via __has_builtin; fallback = ds stores)
// ---------------------------------------------------------------------------
#if defined(__HIP_DEVICE_COMPILE__) && defined(__gfx1250__) && \
    __has_builtin(__builtin_amdgcn_global_load_async_to_lds_b128)
#define HAVE_ASYNC_LDS 1
#endif

#if defined(HAVE_ASYNC_LDS)
__device__ __forceinline__ void wait_async_le2_or_0(bool two) {
#if __has_builtin(__builtin_amdgcn_s_wait_asynccnt)
  if (two) __builtin_amdgcn_s_wait_asynccnt(2);
  else     __builtin_amdgcn_s_wait_asynccnt(0);
#else
  if (two) asm volatile("s_wait_asynccnt 2" ::: "memory");
  else     asm volatile("s_wait_asynccnt 0" ::: "memory");
#endif
}
#endif

// Stage one CHUNK of edge indices (rows then cols) into wave-private LDS.
// Full chunks: one async b128 per lane per array (32 lanes x 16B = 512B each).
__device__ __forceinline__ void stage_chunk(const int* __restrict__ edge, int E,
                                            int cbase, int (*dst)[CHUNK],
                                            int lane) {
  if (cbase + CHUNK <= E) {
#if defined(HAVE_ASYNC_LDS)
    __builtin_amdgcn_global_load_async_to_lds_b128(
        (__attribute__((address_space(1))) v4i_gcc*)(edge + cbase + lane * 4),
        (__attribute__((address_space(3))) v4i_gcc*)&dst[0][lane * 4], 0, 0);
    __builtin_amdgcn_global_load_async_to_lds_b128(
        (__attribute__((address_space(1))) v4i_gcc*)(edge + E + cbase + lane * 4),
        (__attribute__((address_space(3))) v4i_gcc*)&dst[1][lane * 4], 0, 0);
#else
    *(int4*)&dst[0][lane * 4] = *(const int4*)(edge + cbase + lane * 4);
    *(int4*)&dst[1][lane * 4] = *(const int4*)(edge + E + cbase + lane * 4);
#endif
  } else {
    for (int k = lane; k < E - cbase; k += 32) {
      dst[0][k] = edge[cbase + k];
      dst[1][k] = edge[E + cbase + k];
    }
  }
}

// ---------------------------------------------------------------------------
// Scatter: y[row][:] += xs[col][:]  (norm folded into xs / post pass).
// 16 lanes per edge, float4 per lane; edge indices double-buffered through
// LDS via async copies; f32 hardware atomics into L2-resident y.
// (Defined first so the disasm snippet shows this kernel.)
// ---------------------------------------------------------------------------
__global__ __launch_bounds__(TPB) void scatter_kernel(
    const int* __restrict__ edge, int E, const float* __restrict__ xs,
    float* __restrict__ y) {
  __shared__ int lds_idx[TPB / 32][2][2][CHUNK];
  const int lane = threadIdx.x & 31;
  const int wl = threadIdx.x >> 5;
  const int gw = blockIdx.x * (TPB / 32) + wl;
  const int nw = gridDim.x * (TPB / 32);
  const int nchunks = (E + CHUNK - 1) / CHUNK;
  const int half = lane >> 4;  // 2 edges per wave step
  const int sub = lane & 15;   // 16 lanes x float4 = one 64-float row

  int buf = 0;
  if (gw < nchunks) stage_chunk(edge, E, gw * CHUNK, lds_idx[wl][0], lane);
  for (int c = gw; c < nchunks; c += nw) {
    const int cn = c + nw;
    const bool have_next = cn < nchunks;
    if (have_next) stage_chunk(edge, E, cn * CHUNK, lds_idx[wl][buf ^ 1], lane);
#if defined(HAVE_ASYNC_LDS)
    // in-order ASYNCcnt: leaving <=2 outstanding means current chunk landed
    wait_async_le2_or_0(have_next && (cn * CHUNK + CHUNK <= E));
#endif
    const int cbase = c * CHUNK;
    const int rem = (E - cbase < CHUNK) ? (E - cbase) : CHUNK;
    int(*cur)[CHUNK] = lds_idx[wl][buf];
#pragma unroll 4
    for (int it = 0; it < CHUNK / 2; ++it) {
      const int le = it * 2 + half;
      if (le < rem) {
        const int row = cur[0][le];
        const int col = cur[1][le];
        const float4 v = *(const float4*)(xs + (size_t)col * 64 + sub * 4);
        float* dst = y + (size_t)row * 64 + sub * 4;
        unsafeAtomicAdd(dst + 0, v.x);
        unsafeAtomicAdd(dst + 1, v.y);
        unsafeAtomicAdd(dst + 2, v.z);
        unsafeAtomicAdd(dst + 3, v.w);
      }
    }
    buf ^= 1;
  }
}

// ---------------------------------------------------------------------------
// Support kernels
// ---------------------------------------------------------------------------
__global__ __launch_bounds__(TPB) void count_deg_kernel(
    const int* __restrict__ col, int E, int* __restrict__ deg) {
  for (int e = blockIdx.x * TPB + threadIdx.x; e < E; e += gridDim.x * TPB)
    atomicAdd(&deg[col[e]], 1);
}

__global__ __launch_bounds__(TPB) void make_dinv_kernel(float* __restrict__ buf,
                                                        int N) {
  int i = blockIdx.x * TPB + threadIdx.x;
  if (i < N) {
    int d = ((const int*)buf)[i];
    buf[i] = (d > 0) ? (1.0f / sqrtf((float)d)) : 0.0f;
  }
}

// acc = 0.25*emb ; xs = dinv[node]*emb     (float4 per thread; 4 | 64)
__global__ __launch_bounds__(TPB) void init_kernel(
    const float4* __restrict__ ue4, const float4* __restrict__ ie4,
    const float* __restrict__ dinv, float4* __restrict__ xs4,
    float4* __restrict__ acc4, int NU, long nq) {
  for (long q = (long)blockIdx.x * TPB + threadIdx.x; q < nq;
       q += (long)gridDim.x * TPB) {
    int node = (int)(q >> 4);  // 16 float4 per node row
    float4 e = (node < NU) ? ue4[q] : ie4[q - (long)NU * 16];
    float di = dinv[node];
    acc4[q] = make_float4(0.25f * e.x, 0.25f * e.y, 0.25f * e.z, 0.25f * e.w);
    xs4[q] = make_float4(di * e.x, di * e.y, di * e.z, di * e.w);
  }
}

// t = dinv*y ; acc += 0.25*t ; (optionally) xs = dinv*t   (float4 per thread)
__global__ __launch_bounds__(TPB) void post_kernel(
    const float4* __restrict__ y4, const float* __restrict__ dinv,
    float4* __restrict__ acc4, float4* __restrict__ xs4, long nq,
    int write_xs) {
  for (long q = (long)blockIdx.x * TPB + threadIdx.x; q < nq;
       q += (long)gridDim.x * TPB) {
    int node = (int)(q >> 4);
    float di = dinv[node];
    float4 v = y4[q];
    float4 t = make_float4(di * v.x, di * v.y, di * v.z, di * v.w);
    float4 a = acc4[q];
    a.x += 0.25f * t.x; a.y += 0.25f * t.y;
    a.z += 0.25f * t.z; a.w += 0.25f * t.w;
    acc4[q] = a;
    if (write_xs)
      xs4[q] = make_float4(di * t.x, di * t.y, di * t.z, di * t.w);
  }
}

// ---------------------------------------------------------------------------
// Host launcher
// ---------------------------------------------------------------------------
extern "C" void kernel_launch(void* const* d_in, const int* in_sizes, int n_in,
                              void* d_out, int out_size, void* d_ws,
                              size_t ws_size, hipStream_t stream) {
  (void)n_in; (void)out_size; (void)ws_size;
  const int* edge = (const int*)d_in[0];        // [2,E] int32: rows then cols
  const float* uemb = (const float*)d_in[1];    // [NU,64]
  const float* iemb = (const float*)d_in[2];    // [NI,64]
  const int E = in_sizes[0] / 2;
  const int NU = in_sizes[1] / 64;
  const int NI = in_sizes[2] / 64;
  const int N = NU + NI;
  const long nd = (long)N * 64;
  const long nq = nd / 4;  // float4 count

  float* acc = (float*)d_out;                   // [N,64] -> users||items flat
  char* ws = (char*)d_ws;
  float* dinv = (float*)ws;                     // N floats (int deg first)
  size_t off = (((size_t)N * 4) + 255) & ~(size_t)255;
  float* xs = (float*)(ws + off);               // [N,64] pre-scaled features
  float* y = xs + nd;                           // [N,64] scatter target

  auto cdiv = [](long a, long b) { return (int)((a + b - 1) / b); };
  const int blocksE = cdiv(E, TPB);
  const int blocksN = cdiv(N, TPB);
  const int blocksNQ = cdiv(nq, TPB);

  // degree over col, then dinv = deg>0 ? deg^-1/2 : 0  (in place)
  (void)hipMemsetAsync(dinv, 0, (size_t)N * 4, stream);
  count_deg_kernel<<<blocksE, TPB, 0, stream>>>(edge + E, E, (int*)dinv);
  make_dinv_kernel<<<blocksN, TPB, 0, stream>>>(dinv, N);

  // acc = 0.25*emb ; xs = dinv*emb
  init_kernel<<<blocksNQ, TPB, 0, stream>>>(
      (const float4*)uemb, (const float4*)iemb, dinv, (float4*)xs,
      (float4*)acc, NU, nq);

  for (int l = 0; l < 3; ++l) {
    (void)hipMemsetAsync(y, 0, (size_t)nd * 4, stream);
    scatter_kernel<<<SCATTER_BLOCKS, TPB, 0, stream>>>(edge, E, xs, y);
    post_kernel<<<blocksNQ, TPB, 0, stream>>>(
        (const float4*)y, dinv, (float4*)acc, (float4*)xs, nq,
        (l < 2) ? 1 : 0);
  }
}